// GINConv_41094247088185
// MI455X (gfx1250) — compile-verified
//
#include <hip/hip_runtime.h>

typedef __attribute__((ext_vector_type(2))) float v2f;
typedef __attribute__((ext_vector_type(8))) float v8f;

#define IN_DIM 64
#define EMB_DIM 64

// ---------------------------------------------------------------------------
// Kernel 1: h = (1+eps)*x   (float4 streaming), also zero BN-stat accumulators
// ---------------------------------------------------------------------------
__global__ void gin_init(const float* __restrict__ x, const float* __restrict__ epsp,
                         float* __restrict__ h, float* __restrict__ stats, int total4) {
    int gid = blockIdx.x * 256 + threadIdx.x;
    if (blockIdx.x == 0 && threadIdx.x < 128) stats[threadIdx.x] = 0.0f;  // sums[64], sumsq[64]
    if (gid < total4) {
        float s = 1.0f + *epsp;
        float4 v = ((const float4*)x)[gid];
        v.x *= s; v.y *= s; v.z *= s; v.w *= s;
        ((float4*)h)[gid] = v;
    }
}

// ---------------------------------------------------------------------------
// Kernel 2: edge scatter-add: h[dst] += x[src].  16 lanes per edge, float4 each.
// Accumulator (25.6 MB) is L2-resident; atomics resolve in L2.
// ---------------------------------------------------------------------------
__global__ void gin_scatter(const float* __restrict__ x, const int* __restrict__ edge,
                            float* __restrict__ h, int E) {
    int gid = blockIdx.x * 256 + threadIdx.x;
    int e = gid >> 4;
    if (e >= E) return;
    int j = (gid & 15) * 4;
    int src = edge[e];        // edge_index[0][e]
    int dst = edge[E + e];    // edge_index[1][e]
    float4 v = *(const float4*)(x + (size_t)src * IN_DIM + j);
    float* p = h + (size_t)dst * IN_DIM + j;
    atomicAdd(p + 0, v.x);
    atomicAdd(p + 1, v.y);
    atomicAdd(p + 2, v.z);
    atomicAdd(p + 3, v.w);
}

// ---------------------------------------------------------------------------
// Kernel 3: h1 = h @ W1 + b1 via V_WMMA_F32_16X16X4_F32, with fused BN-stat
// reduction (per-column sum and sum of squares) through LDS then global atomics.
// Block = 256 threads = 8 waves; each wave owns a 16x16 output tile
// (2 row-tiles x 4 col-tiles -> 32 rows x 64 cols per block).
// ---------------------------------------------------------------------------
__global__ void __launch_bounds__(256) gin_gemm1(const float* __restrict__ H,
                                                 const float* __restrict__ W,
                                                 const float* __restrict__ bias,
                                                 float* __restrict__ H1,
                                                 float* __restrict__ stats) {
    __shared__ float s_red[128];  // [0:64) col sums, [64:128) col sumsq
    int tid = threadIdx.x;
    if (tid < 128) s_red[tid] = 0.0f;
    __syncthreads();

    int wave = tid >> 5;
    int lane = tid & 31;
    int half = lane >> 4;   // lane 0-15 -> 0, lane 16-31 -> 1
    int l16  = lane & 15;
    int m0 = blockIdx.x * 32 + (wave >> 2) * 16;
    int n0 = (wave & 3) * 16;

    // A layout (16x4 f32): v0 = K0 (lanes 0-15) / K2 (lanes 16-31); v1 = K1 / K3.
    const float* arow = H + (size_t)(m0 + l16) * IN_DIM + half * 2;
    __builtin_prefetch(arow, 0, 3);

    v8f acc = {};
    for (int k = 0; k < IN_DIM; k += 4) {
        v2f a = *(const v2f*)(arow + k);
        int kk = k + half * 2;
        v2f b;
        b.x = W[(size_t)kk * EMB_DIM + n0 + l16];
        b.y = W[(size_t)(kk + 1) * EMB_DIM + n0 + l16];
        acc = __builtin_amdgcn_wmma_f32_16x16x4_f32(
            /*neg_a=*/false, a, /*neg_b=*/false, b,
            /*c_mod=*/(short)0, acc, /*reuse_a=*/false, /*reuse_b=*/false);
    }

    // C/D layout: VGPR r -> row m0+r (lanes 0-15) / m0+r+8 (lanes 16-31), col n0+l16
    int col = n0 + l16;
    float bv = bias[col];
    float lsum = 0.0f, lsq = 0.0f;
#pragma unroll
    for (int r = 0; r < 8; ++r) {
        int row = m0 + half * 8 + r;
        float v = acc[r] + bv;
        H1[(size_t)row * EMB_DIM + col] = v;
        lsum += v;
        lsq += v * v;
    }
    atomicAdd(&s_red[col], lsum);        // ds_add_f32
    atomicAdd(&s_red[64 + col], lsq);
    __syncthreads();
    if (tid < 128) atomicAdd(&stats[tid], s_red[tid]);  // one global atomic/col/block
}

// ---------------------------------------------------------------------------
// Kernel 4: fold BN stats into per-feature scale/shift (64 lanes of work).
// ---------------------------------------------------------------------------
__global__ void gin_bnprep(const float* __restrict__ gamma, const float* __restrict__ beta,
                           float* __restrict__ stats, float invN) {
    int t = threadIdx.x;
    if (t < 64) {
        float mean = stats[t] * invN;
        float var  = stats[64 + t] * invN - mean * mean;   // biased variance
        float rstd = rsqrtf(var + 1e-5f);
        float sc = gamma[t] * rstd;
        stats[128 + t] = sc;                  // scale
        stats[192 + t] = beta[t] - mean * sc; // shift
    }
}

// ---------------------------------------------------------------------------
// Kernel 5: out = relu(h1*scale + shift) @ W2 + b2.  BN-apply + ReLU fused into
// the WMMA A-operand load.
// ---------------------------------------------------------------------------
__global__ void __launch_bounds__(256) gin_gemm2(const float* __restrict__ H1,
                                                 const float* __restrict__ W,
                                                 const float* __restrict__ bias,
                                                 const float* __restrict__ stats,
                                                 float* __restrict__ out) {
    int tid = threadIdx.x;
    int wave = tid >> 5;
    int lane = tid & 31;
    int half = lane >> 4;
    int l16  = lane & 15;
    int m0 = blockIdx.x * 32 + (wave >> 2) * 16;
    int n0 = (wave & 3) * 16;

    const float* scale = stats + 128;
    const float* shift = stats + 192;
    const float* arow = H1 + (size_t)(m0 + l16) * EMB_DIM + half * 2;
    __builtin_prefetch(arow, 0, 3);

    v8f acc = {};
    for (int k = 0; k < EMB_DIM; k += 4) {
        int kk = k + half * 2;
        v2f raw = *(const v2f*)(arow + k);
        v2f a;
        a.x = fmaxf(raw.x * scale[kk]     + shift[kk],     0.0f);
        a.y = fmaxf(raw.y * scale[kk + 1] + shift[kk + 1], 0.0f);
        v2f b;
        b.x = W[(size_t)kk * EMB_DIM + n0 + l16];
        b.y = W[(size_t)(kk + 1) * EMB_DIM + n0 + l16];
        acc = __builtin_amdgcn_wmma_f32_16x16x4_f32(
            false, a, false, b, (short)0, acc, false, false);
    }

    int col = n0 + l16;
    float bv = bias[col];
#pragma unroll
    for (int r = 0; r < 8; ++r) {
        int row = m0 + half * 8 + r;
        out[(size_t)row * EMB_DIM + col] = acc[r] + bv;
    }
}

// ---------------------------------------------------------------------------
extern "C" void kernel_launch(void* const* d_in, const int* in_sizes, int n_in,
                              void* d_out, int out_size, void* d_ws, size_t ws_size,
                              hipStream_t stream) {
    const float* x     = (const float*)d_in[0];
    const int*   edge  = (const int*)  d_in[1];
    const float* epsp  = (const float*)d_in[2];
    const float* W1    = (const float*)d_in[3];
    const float* b1    = (const float*)d_in[4];
    const float* gamma = (const float*)d_in[5];
    const float* beta  = (const float*)d_in[6];
    const float* W2    = (const float*)d_in[7];
    const float* b2    = (const float*)d_in[8];
    float* out = (float*)d_out;

    int N = in_sizes[0] / IN_DIM;   // 100000
    int E = in_sizes[1] / 2;        // 1600000

    float* ws    = (float*)d_ws;
    float* h     = ws;                         // N*64 floats
    float* h1    = ws + (size_t)N * IN_DIM;    // N*64 floats
    float* stats = h1 + (size_t)N * EMB_DIM;   // sums[64], sumsq[64], scale[64], shift[64]

    int total4 = N * (IN_DIM / 4);
    gin_init<<<(total4 + 255) / 256, 256, 0, stream>>>(x, epsp, h, stats, total4);

    long long sthreads = (long long)E * 16;
    gin_scatter<<<(unsigned)((sthreads + 255) / 256), 256, 0, stream>>>(x, edge, h, E);

    gin_gemm1<<<(N + 31) / 32, 256, 0, stream>>>(h, W1, b1, h1, stats);

    gin_bnprep<<<1, 64, 0, stream>>>(gamma, beta, stats, 1.0f / (float)N);

    gin_gemm2<<<(N + 31) / 32, 256, 0, stream>>>(h1, W2, b2, stats, out);
}